// Network_51204600103594
// MI455X (gfx1250) — compile-verified
//
#include <hip/hip_runtime.h>
#include <hip/hip_bf16.h>

#define NN 512
#define LL 2048
#define BB 256
#define KK 256

typedef __attribute__((ext_vector_type(2))) float v2f;
typedef __attribute__((ext_vector_type(8))) float v8f;

// ---------------------------------------------------------------------------
// Kernel 1: closed-form exp(i*B2) for each 2x2 Hermitian block.
// U stored as 8 floats per layer: u00r,u00i, u01r,u01i, u10r,u10i, u11r,u11i
// ---------------------------------------------------------------------------
__global__ void k_givens_u(const float* __restrict__ params, float* __restrict__ U) {
    int l = blockIdx.x * blockDim.x + threadIdx.x;
    if (l >= LL) return;
    float bii = params[l * 4 + 0];
    float bjj = params[l * 4 + 1];
    float br  = params[l * 4 + 2];
    float bi  = params[l * 4 + 3];
    float c  = 0.5f * (bii + bjj);
    float th = 0.5f * (bii - bjj);
    float r  = sqrtf(th * th + br * br + bi * bi);
    float sinc = (r > 0.f) ? (sinf(r) / r) : 1.0f;
    float cosr = cosf(r);
    float phr = cosf(c), phi = sinf(c);
    // u00 = ph * (cosr + i*sinc*th)
    float t_re = cosr, t_im = sinc * th;
    float u00r = phr * t_re - phi * t_im;
    float u00i = phr * t_im + phi * t_re;
    // u01 = ph * (i*sinc*(br + i*bi)) = ph * (-sinc*bi + i*sinc*br)
    t_re = -sinc * bi; t_im = sinc * br;
    float u01r = phr * t_re - phi * t_im;
    float u01i = phr * t_im + phi * t_re;
    // u10 = ph * (i*sinc*(br - i*bi)) = ph * (sinc*bi + i*sinc*br)
    t_re = sinc * bi; t_im = sinc * br;
    float u10r = phr * t_re - phi * t_im;
    float u10i = phr * t_im + phi * t_re;
    // u11 = ph * (cosr - i*sinc*th)
    t_re = cosr; t_im = -sinc * th;
    float u11r = phr * t_re - phi * t_im;
    float u11i = phr * t_im + phi * t_re;
    float* u = U + l * 8;
    u[0] = u00r; u[1] = u00i; u[2] = u01r; u[3] = u01i;
    u[4] = u10r; u[5] = u10i; u[6] = u11r; u[7] = u11i;
}

// ---------------------------------------------------------------------------
// Kernel 2a: V = I (interleaved complex, row-major NxN)
// ---------------------------------------------------------------------------
__global__ void k_init_v(float* __restrict__ V) {
    int idx = blockIdx.x * blockDim.x + threadIdx.x;  // < N*N
    int row = idx >> 9, col = idx & (NN - 1);
    V[idx * 2]     = (row == col) ? 1.0f : 0.0f;
    V[idx * 2 + 1] = 0.0f;
}

// ---------------------------------------------------------------------------
// Kernel 2b: apply one sweep of disjoint Givens column rotations.
// blockIdx.y = pair within sweep, row = blockIdx.x*256 + tid.
// V[:, i] <- V[:,i]*u00 + V[:,j]*u10 ; V[:, j] <- V[:,i]*u01 + V[:,j]*u11
// ---------------------------------------------------------------------------
__global__ void k_sweep(const float* __restrict__ U, const int* __restrict__ conn,
                        float* __restrict__ V, int l0) {
    int layer = l0 + blockIdx.y;
    int row = blockIdx.x * blockDim.x + threadIdx.x;
    int ci = conn[layer * 2 + 0];
    int cj = conn[layer * 2 + 1];
    const float* u = U + layer * 8;
    float u00r = u[0], u00i = u[1], u01r = u[2], u01i = u[3];
    float u10r = u[4], u10i = u[5], u11r = u[6], u11i = u[7];
    float* vi = V + ((size_t)row * NN + ci) * 2;
    float* vj = V + ((size_t)row * NN + cj) * 2;
    float ar = vi[0], ai = vi[1], brr = vj[0], bii = vj[1];
    float nir = ar * u00r - ai * u00i + brr * u10r - bii * u10i;
    float nii = ar * u00i + ai * u00r + brr * u10i + bii * u10r;
    float njr = ar * u01r - ai * u01i + brr * u11r - bii * u11i;
    float nji = ar * u01i + ai * u01r + brr * u11i + bii * u11r;
    vi[0] = nir; vi[1] = nii;
    vj[0] = njr; vj[1] = nji;
}

// ---------------------------------------------------------------------------
// Kernel 3: per-batch ascending index lists of the k-hot masks.
// t < BB -> scan x into rows; t >= BB -> scan y into cols.
// (stable argsort(-khot) == ascending indices of ones)
// ---------------------------------------------------------------------------
__global__ void k_select(const int* __restrict__ x, const int* __restrict__ y,
                         int* __restrict__ rows, int* __restrict__ cols) {
    int t = blockIdx.x * blockDim.x + threadIdx.x;
    if (t >= 2 * BB) return;
    int b = t & (BB - 1);
    const int* src = (t < BB) ? x : y;
    int* dst = (t < BB) ? rows : cols;
    int cnt = 0;
    for (int n = 0; n < NN && cnt < KK; ++n) {
        if (src[b * NN + n] != 0) { dst[b * KK + cnt] = n; ++cnt; }
    }
    for (; cnt < KK; ++cnt) dst[b * KK + cnt] = 0;
}

// ---------------------------------------------------------------------------
// Kernel 4: gather V_tild[b] = V[rows[b][i], cols[b][j]] into a ws slot.
// ---------------------------------------------------------------------------
__global__ void k_gather(const float* __restrict__ V, const int* __restrict__ rows,
                         const int* __restrict__ cols, float* __restrict__ mats,
                         int batch0) {
    int b = batch0 + blockIdx.y;
    int idx = blockIdx.x * blockDim.x + threadIdx.x;  // < K*K
    int i = idx >> 8, j = idx & (KK - 1);
    int r = rows[b * KK + i];
    int c = cols[b * KK + j];
    float* m = mats + (size_t)blockIdx.y * KK * KK * 2 + (size_t)idx * 2;
    const float* v = V + ((size_t)r * NN + c) * 2;
    m[0] = v[0];
    m[1] = v[1];
}

// ---------------------------------------------------------------------------
// Kernel 5: blocked complex LU (partial pivoting) + det, one workgroup/matrix.
// 256 threads = 8 wave32. Trailing Schur updates use V_WMMA_F32_16X16X4_F32:
// complex GEMM = 4 real 16x16x16 GEMMs, each as 4 chained 16x16x4 WMMAs.
// Subtraction folded by negating A operands in VALU (f32 NEG only has CNeg).
// ---------------------------------------------------------------------------
__global__ void __launch_bounds__(256) k_lu_det(float* __restrict__ mats,
                                                float* __restrict__ out,
                                                int batch0, int out_cap) {
    const int tid  = threadIdx.x;
    const int lane = tid & 31;
    const int wave = tid >> 5;
    float* M = mats + (size_t)blockIdx.x * KK * KK * 2;

    __shared__ float red[256];
    __shared__ int   redi[256];
    __shared__ float Bs[16 * 240 * 2];   // staged U12 block (<=16x240 complex)
    __shared__ float sdet[2];

    if (tid == 0) { sdet[0] = 1.0f; sdet[1] = 0.0f; }
    __syncthreads();

    for (int kb = 0; kb < 16; ++kb) {
        const int j0 = kb * 16;
        const int j1 = j0 + 16;

        // ---- panel factorization of columns [j0, j1) over rows [j0, K) ----
        for (int j = j0; j < j1; ++j) {
            // partial pivot: argmax |M[i][j]|^2, i >= j
            float best = -1.0f; int bidx = j;
            for (int i2 = j + tid; i2 < KK; i2 += 256) {
                float re = M[(i2 * KK + j) * 2];
                float im = M[(i2 * KK + j) * 2 + 1];
                float a = re * re + im * im;
                if (a > best) { best = a; bidx = i2; }
            }
            red[tid] = best; redi[tid] = bidx;
            __syncthreads();
            for (int s = 128; s > 0; s >>= 1) {
                if (tid < s && red[tid + s] > red[tid]) {
                    red[tid] = red[tid + s]; redi[tid] = redi[tid + s];
                }
                __syncthreads();
            }
            int piv = redi[0];
            __syncthreads();
            if (piv != j) {  // swap full rows j <-> piv, one column per thread
                int c2 = tid;
                float tr = M[(j * KK + c2) * 2];
                float ti = M[(j * KK + c2) * 2 + 1];
                M[(j * KK + c2) * 2]       = M[(piv * KK + c2) * 2];
                M[(j * KK + c2) * 2 + 1]   = M[(piv * KK + c2) * 2 + 1];
                M[(piv * KK + c2) * 2]     = tr;
                M[(piv * KK + c2) * 2 + 1] = ti;
            }
            __syncthreads();
            float pre = M[(j * KK + j) * 2];
            float pim = M[(j * KK + j) * 2 + 1];
            float n2  = pre * pre + pim * pim;
            float ivr = (n2 > 0.f) ? ( pre / n2) : 0.f;
            float ivi = (n2 > 0.f) ? (-pim / n2) : 0.f;
            if (tid == 0) {  // det *= pivot (with swap sign)
                float sr = (piv != j) ? -pre : pre;
                float si = (piv != j) ? -pim : pim;
                float dr = sdet[0] * sr - sdet[1] * si;
                float di = sdet[0] * si + sdet[1] * sr;
                sdet[0] = dr; sdet[1] = di;
            }
            // multipliers: M[i][j] *= 1/pivot  (i > j)
            for (int i2 = j + 1 + tid; i2 < KK; i2 += 256) {
                float ar2 = M[(i2 * KK + j) * 2];
                float ai2 = M[(i2 * KK + j) * 2 + 1];
                M[(i2 * KK + j) * 2]     = ar2 * ivr - ai2 * ivi;
                M[(i2 * KK + j) * 2 + 1] = ar2 * ivi + ai2 * ivr;
            }
            __syncthreads();
            // rank-1 update restricted to remaining panel columns
            for (int i2 = j + 1 + tid; i2 < KK; i2 += 256) {
                float lr = M[(i2 * KK + j) * 2];
                float li = M[(i2 * KK + j) * 2 + 1];
                for (int jj = j + 1; jj < j1; ++jj) {
                    float ur = M[(j * KK + jj) * 2];
                    float ui = M[(j * KK + jj) * 2 + 1];
                    M[(i2 * KK + jj) * 2]     -= lr * ur - li * ui;
                    M[(i2 * KK + jj) * 2 + 1] -= lr * ui + li * ur;
                }
            }
            __syncthreads();
        }

        const int ncols = KK - j1;
        if (ncols <= 0) continue;

        // ---- triangular solve U12 = L11^{-1} A12 : one column per thread ----
        if (tid < ncols) {
            int c2 = j1 + tid;
            float ar2[16], ai2[16];
#pragma unroll
            for (int t = 0; t < 16; ++t) {
                ar2[t] = M[((j0 + t) * KK + c2) * 2];
                ai2[t] = M[((j0 + t) * KK + c2) * 2 + 1];
            }
#pragma unroll
            for (int jr = 1; jr < 16; ++jr) {
#pragma unroll
                for (int t = 0; t < jr; ++t) {
                    float lr = M[((j0 + jr) * KK + j0 + t) * 2];
                    float li = M[((j0 + jr) * KK + j0 + t) * 2 + 1];
                    ar2[jr] -= lr * ar2[t] - li * ai2[t];
                    ai2[jr] -= lr * ai2[t] + li * ar2[t];
                }
            }
#pragma unroll
            for (int t = 0; t < 16; ++t) {
                M[((j0 + t) * KK + c2) * 2]     = ar2[t];
                M[((j0 + t) * KK + c2) * 2 + 1] = ai2[t];
            }
        }
        __syncthreads();

        // ---- stage U12 block into LDS (B operand) ----
        for (int idx = tid; idx < 16 * ncols; idx += 256) {
            int kr = idx / ncols, cc = idx - kr * ncols;
            Bs[(kr * 240 + cc) * 2]     = M[((j0 + kr) * KK + j1 + cc) * 2];
            Bs[(kr * 240 + cc) * 2 + 1] = M[((j0 + kr) * KK + j1 + cc) * 2 + 1];
        }
        __syncthreads();

        // ---- trailing update A22 -= L21 * U12 via WMMA f32 16x16x4 ----
        const int nt = ncols >> 4;
        for (int t = wave; t < nt * nt; t += 8) {
            int tr = t / nt, tc = t - tr * nt;
            int r0    = j1 + tr * 16;
            int c0rel = tc * 16;
            int c0    = j1 + c0rel;
            // operand lane mapping (wave32):
            // A 16x4: lanes 0-15 rows, vgpr0={K0|K2}, vgpr1={K1|K3}
            int mrow  = r0 + (lane & 15);
            int kloc  = (lane >> 4) << 1;
            int crow0 = r0 + ((lane >> 4) << 3);
            int ccol  = c0 + (lane & 15);
            int ccrel = c0rel + (lane & 15);
            v8f accR, accI;
#pragma unroll
            for (int v = 0; v < 8; ++v) {
                accR[v] = M[((crow0 + v) * KK + ccol) * 2];
                accI[v] = M[((crow0 + v) * KK + ccol) * 2 + 1];
            }
#pragma unroll
            for (int s = 0; s < 4; ++s) {
                int kk = 4 * s + kloc;                  // k offset within panel
                v2f Ar, Ai, Br, Bi;
                Ar[0] = M[(mrow * KK + j0 + kk) * 2];
                Ai[0] = M[(mrow * KK + j0 + kk) * 2 + 1];
                Ar[1] = M[(mrow * KK + j0 + kk + 1) * 2];
                Ai[1] = M[(mrow * KK + j0 + kk + 1) * 2 + 1];
                Br[0] = Bs[(kk * 240 + ccrel) * 2];
                Bi[0] = Bs[(kk * 240 + ccrel) * 2 + 1];
                Br[1] = Bs[((kk + 1) * 240 + ccrel) * 2];
                Bi[1] = Bs[((kk + 1) * 240 + ccrel) * 2 + 1];
                v2f nAr = -Ar, nAi = -Ai;
                // Cr += (-Ar)*Br + Ai*Bi ; Ci += (-Ar)*Bi + (-Ai)*Br
                accR = __builtin_amdgcn_wmma_f32_16x16x4_f32(false, nAr, false, Br, (short)0, accR, false, false);
                accR = __builtin_amdgcn_wmma_f32_16x16x4_f32(false, Ai,  false, Bi, (short)0, accR, false, false);
                accI = __builtin_amdgcn_wmma_f32_16x16x4_f32(false, nAr, false, Bi, (short)0, accI, false, false);
                accI = __builtin_amdgcn_wmma_f32_16x16x4_f32(false, nAi, false, Br, (short)0, accI, false, false);
            }
#pragma unroll
            for (int v = 0; v < 8; ++v) {
                M[((crow0 + v) * KK + ccol) * 2]     = accR[v];
                M[((crow0 + v) * KK + ccol) * 2 + 1] = accI[v];
            }
        }
        __syncthreads();
    }

    if (tid == 0) {
        int b = batch0 + (int)blockIdx.x;
        if (2 * b < out_cap)     out[2 * b]     = sdet[0];
        if (2 * b + 1 < out_cap) out[2 * b + 1] = sdet[1];
    }
}

// ---------------------------------------------------------------------------
// Host launcher. Workspace layout:
//   [0,64K)            U        (L*8 floats)
//   [64K, 64K+2M)      V        (N*N*2 floats, interleaved complex)
//   next 256K          rows     (B*K ints)
//   next 256K          cols     (B*K ints)
//   rest               matrices (slots * K*K*2 floats), chunked if needed
// ---------------------------------------------------------------------------
extern "C" void kernel_launch(void* const* d_in, const int* in_sizes, int n_in,
                              void* d_out, int out_size, void* d_ws, size_t ws_size,
                              hipStream_t stream) {
    const float* params = (const float*)d_in[0];
    const int*   conn   = (const int*)d_in[1];
    const int*   y      = (const int*)d_in[2];
    const int*   x      = (const int*)d_in[3];
    float*       out    = (float*)d_out;

    char* ws = (char*)d_ws;
    const size_t offU    = 0;
    const size_t offV    = offU + (size_t)LL * 8 * sizeof(float);
    const size_t offRows = offV + (size_t)NN * NN * 2 * sizeof(float);
    const size_t offCols = offRows + (size_t)BB * KK * sizeof(int);
    const size_t offMats = offCols + (size_t)BB * KK * sizeof(int);
    float* U    = (float*)(ws + offU);
    float* V    = (float*)(ws + offV);
    int*   rows = (int*)(ws + offRows);
    int*   cols = (int*)(ws + offCols);
    float* mats = (float*)(ws + offMats);

    const size_t matBytes = (size_t)KK * KK * 2 * sizeof(float);
    size_t avail = (ws_size > offMats) ? (ws_size - offMats) : 0;
    int slots = (int)(avail / matBytes);
    if (slots > BB) slots = BB;
    if (slots < 1)  slots = 1;  // best effort

    // 1) 2x2 unitaries
    k_givens_u<<<LL / 256, 256, 0, stream>>>(params, U);
    // 2) V = I, then 9 disjoint Givens sweeps (conn tiles 256 even + 255 odd pairs)
    k_init_v<<<(NN * NN) / 256, 256, 0, stream>>>(V);
    int l0 = 0;
    while (l0 < LL) {
        int pos = l0 % 511;
        int run = (pos < 256) ? (256 - pos) : (511 - pos);
        int l1 = l0 + run;
        if (l1 > LL) l1 = LL;
        dim3 g(NN / 256, l1 - l0);
        k_sweep<<<g, 256, 0, stream>>>(U, conn, V, l0);
        l0 = l1;
    }
    // 3) row/col index lists
    k_select<<<2, 256, 0, stream>>>(x, y, rows, cols);
    // 4+5) gather submatrices and LU-det them, chunked to fit workspace
    for (int c = 0; c < BB; c += slots) {
        int nb = (BB - c < slots) ? (BB - c) : slots;
        dim3 gg((KK * KK) / 256, nb);
        k_gather<<<gg, 256, 0, stream>>>(V, rows, cols, mats, c);
        k_lu_det<<<nb, 256, 0, stream>>>(mats, out, c, out_size);
    }
}